// HealpixSmoothingLayer_63419487092962
// MI455X (gfx1250) — compile-verified
//
#include <hip/hip_runtime.h>
#include <hip/hip_bf16.h>
#include <stdint.h>

// ---------------------------------------------------------------------------
// HealpixSmoothingLayer on MI455X (gfx1250)
//   out[b,n,c] = sum_k ( kernel[n,k] / rowsum[inds[n,k]] ) * x[b, inds[n,k], c]
// Memory/gather bound (AI ~0.5 flop/byte) -> use the CDNA5 data path:
//   TDM tensor_load_to_lds (with D# pad feature for bank-conflict-free LDS
//   rows), wave32 occupancy, L2-resident gathers, nontemporal coalesced
//   stores. WMMA is structurally inapplicable (per-row gathered B operand).
// ---------------------------------------------------------------------------

#define KNB   32   // MAX_NEIGHBORS
#define NBATCH 8   // BATCH
#define BLK   256  // pixels (= threads) per block
#define ROWP  33   // padded LDS row stride in dwords (33 odd -> conflict-free)

typedef __attribute__((ext_vector_type(4))) float        f4;
typedef __attribute__((ext_vector_type(4))) unsigned int v4u;
typedef __attribute__((ext_vector_type(4))) int          v4i;
typedef __attribute__((ext_vector_type(8))) int          v8i;

#if __has_builtin(__builtin_amdgcn_tensor_load_to_lds) && \
    __has_builtin(__builtin_amdgcn_s_wait_tensorcnt)
#define HAVE_TDM 1
#else
#define HAVE_TDM 0
#endif

#if HAVE_TDM
// Issue one TDM 2-D tile load: tile_w (=32) dwords x tile_h rows, row stride =
// tile_w elements, tensor_dim1 = rows_avail (hardware zero-fills OOB rows).
// D# pad feature inserts 1 dword after every 32 dwords -> ROWP=33 LDS layout.
__device__ __forceinline__ void tdm_load_tile_2d(unsigned lds_byte_addr,
                                                 const void* gsrc,
                                                 unsigned tile_w,
                                                 unsigned tile_h,
                                                 unsigned rows_avail) {
  const uint64_t ga = (uint64_t)(uintptr_t)gsrc;

  // ---- D# group 0 (128b): count=1 | lds_addr | global_addr | type=2 ----
  v4u g0;
  g0.x = 1u;                                           // count = 1 valid desc
  g0.y = lds_byte_addr;                                // LDS byte address
  g0.z = (unsigned)(ga & 0xFFFFFFFFull);               // global_addr[31:0]
  g0.w = (unsigned)((ga >> 32) & 0x01FFFFFFu)          // global_addr[56:32]
       | (2u << 30);                                   // type = 2 ("image")

  // ---- D# group 1 (256b) ----
  // dw0: data_size=2 (4B), pad_enable=1, pad_interval code 4 (=32 dwords),
  //      pad_amount code 0 (=1 dword)
  v8i g1;
  g1[0] = (int)((2u << 16) | (1u << 20) | (4u << 22));
  g1[1] = (int)((tile_w & 0xFFFFu) << 16);             // tensor_dim0[15:0]
  g1[2] = (int)(((tile_w >> 16) & 0xFFFFu)             // tensor_dim0[31:16]
              | ((rows_avail & 0xFFFFu) << 16));       // tensor_dim1[15:0]
  g1[3] = (int)(((rows_avail >> 16) & 0xFFFFu)         // tensor_dim1[31:16]
              | ((tile_w & 0xFFFFu) << 16));           // tile_dim0
  g1[4] = (int)(tile_h & 0xFFFFu);                     // tile_dim1 (tile_dim2=0)
  g1[5] = (int)tile_w;                                 // tensor_dim0_stride[31:0]
  g1[6] = 0;                                           // stride hi / dim1_stride lo
  g1[7] = 0;

#if defined(__clang_major__) && (__clang_major__ >= 23)
  __builtin_amdgcn_tensor_load_to_lds(g0, g1, (v4i)0, (v4i)0, (v8i)0, 0);
#else
  __builtin_amdgcn_tensor_load_to_lds(g0, g1, (v4i)0, (v4i)0, 0);
#endif
}
#endif // HAVE_TDM

// ---------------------------------------------------------------------------
// Main kernel (placed first so the disasm snippet shows the TDM sequence):
// gather-weighted reduction, one pixel per lane.
// ---------------------------------------------------------------------------
__global__ __launch_bounds__(BLK) void healpix_smooth_kernel(
    const float* __restrict__ x, const float* __restrict__ kern,
    const int* __restrict__ inds, const float* __restrict__ invsum,
    float* __restrict__ out, int npix) {
  __shared__ float sK[BLK * ROWP];  // 33 KB, bank-conflict-free rows
  __shared__ int   sI[BLK * ROWP];  // 33 KB

  const int tid = threadIdx.x;
  const int n0  = blockIdx.x * BLK;

#if HAVE_TDM
  // Wave 0 DMAs both tiles into LDS via the Tensor Data Mover, then waits on
  // TENSORcnt; everyone else just hits the workgroup barrier.
  if (tid < 32) {
    tdm_load_tile_2d((unsigned)(uintptr_t)&sK[0],
                     kern + (size_t)n0 * KNB, KNB, BLK, (unsigned)(npix - n0));
    tdm_load_tile_2d((unsigned)(uintptr_t)&sI[0],
                     inds + (size_t)n0 * KNB, KNB, BLK, (unsigned)(npix - n0));
    __builtin_amdgcn_s_wait_tensorcnt(0);
  }
  __syncthreads();
#else
  // Fallback: cooperative coalesced copy into the same padded layout.
  for (int i = tid; i < BLK * KNB; i += BLK) {
    const int p = i / KNB, j = i % KNB;
    const int g = n0 + p;
    sK[p * ROWP + j] = (g < npix) ? kern[(size_t)g * KNB + j] : 0.0f;
    sI[p * ROWP + j] = (g < npix) ? inds[(size_t)g * KNB + j] : 0;
  }
  __syncthreads();
#endif

  const int n = n0 + tid;
  if (n >= npix) return;

  f4 acc[NBATCH];
#pragma unroll
  for (int b = 0; b < NBATCH; ++b) acc[b] = (f4)0.0f;

  const float* sKr = &sK[tid * ROWP];
  const int*   sIr = &sI[tid * ROWP];

#pragma unroll 8
  for (int k = 0; k < KNB; ++k) {
    const int   idx = sIr[k];
    const float wk  = sKr[k] * invsum[idx];          // 4B gather, L2-resident
    const f4*   xp  = (const f4*)(x + (size_t)idx * 4);
#pragma unroll
    for (int b = 0; b < NBATCH; ++b) {
      acc[b] += wk * xp[(size_t)b * (size_t)npix];   // 16B gathers of x[b,idx,:]
    }
  }

#pragma unroll
  for (int b = 0; b < NBATCH; ++b) {
    f4* op = (f4*)(out + ((size_t)b * (size_t)npix + (size_t)n) * 4);
    __builtin_nontemporal_store(acc[b], op);         // streaming store, keep L2 for x
  }
}

// ---------------------------------------------------------------------------
// Kernel 1: inv_rowsum[n] = 1 / sum_k kernel[n,k]
// ---------------------------------------------------------------------------
__global__ __launch_bounds__(BLK) void healpix_invrowsum_kernel(
    const float* __restrict__ kern, float* __restrict__ inv, int npix) {
  const int n = blockIdx.x * BLK + threadIdx.x;
  if (n >= npix) return;
  const f4* r = (const f4*)(kern + (size_t)n * KNB);
  f4 s4 = r[0];
#pragma unroll
  for (int j = 1; j < KNB / 4; ++j) s4 += r[j];
  const float s = (s4.x + s4.y) + (s4.z + s4.w);
  inv[n] = 1.0f / s;
}

// ---------------------------------------------------------------------------
extern "C" void kernel_launch(void* const* d_in, const int* in_sizes, int n_in,
                              void* d_out, int out_size, void* d_ws, size_t ws_size,
                              hipStream_t stream) {
  const float* x    = (const float*)d_in[0];   // (B, N_PIX, C) f32
  const float* kern = (const float*)d_in[1];   // (N_PIX, 32)   f32
  const int*   inds = (const int*)d_in[2];     // (N_PIX, 32)   int
  float*       out  = (float*)d_out;           // (B, N_PIX, C) f32
  float*       inv  = (float*)d_ws;            // N_PIX f32 scratch

  const int npix   = in_sizes[1] / KNB;
  const int blocks = (npix + BLK - 1) / BLK;

  healpix_invrowsum_kernel<<<blocks, BLK, 0, stream>>>(kern, inv, npix);
  healpix_smooth_kernel<<<blocks, BLK, 0, stream>>>(x, kern, inds, inv, out, npix);
}